// ODEFunc_89996744720506
// MI455X (gfx1250) — compile-verified
//
#include <hip/hip_runtime.h>

typedef float v2f __attribute__((ext_vector_type(2)));
typedef float v8f __attribute__((ext_vector_type(8)));

// global-address-space pointer types (force global_load/global_store, not flat)
typedef __attribute__((address_space(1))) const float gcfloat;
typedef __attribute__((address_space(1))) float       gfloat;
typedef __attribute__((address_space(1))) const v2f   gcv2f;
typedef __attribute__((address_space(1))) v2f         gv2f;

__device__ __forceinline__ gcfloat* as_g(const float* p) {
    return (gcfloat*)(unsigned long long)p;
}
__device__ __forceinline__ gfloat* as_gm(float* p) {
    return (gfloat*)(unsigned long long)p;
}

#define TPAD 40                 // LDS row stride (floats): 2*40 % 64 == 16 -> conflict-free halves
#define TOK  32                 // tokens per workgroup
static constexpr long NTOK = 2048L * 128L;   // total tokens (B*S)

static constexpr int KT_CTX = 33;   // ceil(130/4): context K tiles (padded to 132)
static constexpr int KT_H   = 64;   // 256/4: hidden K tiles
static constexpr int KT_Z   = 1;    // ceil(2/4): input K tiles
static constexpr int MT_H   = 16;   // 256/16 output tiles
static constexpr int MT_O   = 1;    // ceil(2/16) output tiles

__device__ __forceinline__ v8f wmma_f32(v2f a, v2f b, v8f c) {
    // D = A(16x4 f32) * B(4x16 f32) + C(16x16 f32)
    return __builtin_amdgcn_wmma_f32_16x16x4_f32(false, a, false, b, (short)0, c, false, false);
}

__device__ __forceinline__ float sigmoidf(float x) { return 1.0f / (1.0f + __expf(-x)); }
__device__ __forceinline__ float softplusf(float x) {
    // stable softplus; log argument is in (1,2] -> __logf is fully accurate there
    return fmaxf(x, 0.0f) + __logf(1.0f + __expf(-fabsf(x)));
}
__device__ __forceinline__ v8f zero8() {
    v8f z = {0.f,0.f,0.f,0.f,0.f,0.f,0.f,0.f};
    return z;
}

// ---------------- weight repack: row-major W[DOUT][DIN] -> A-tile stream ----------------
// packed[(mt*KT + kt)*64 + lane*2 + c] = W[mt*16 + (lane&15)][kt*4 + 2*(lane>>4) + c], 0-padded.
__global__ __launch_bounds__(256)
void pack_kernel(const float* __restrict__ src, float* __restrict__ dst,
                 int DOUT, int DIN, int KT, int total) {
    int e = blockIdx.x * 256 + threadIdx.x;
    if (e >= total) return;
    int tile = e >> 6;
    int lane = (e >> 1) & 31;
    int comp = e & 1;
    int lo = lane & 15, hi = lane >> 4;
    int kt = tile % KT, mt = tile / KT;
    int row = mt * 16 + lo;
    int col = kt * 4 + 2 * hi + comp;
    float v = (row < DOUT && col < DIN) ? as_g(src)[row * DIN + col] : 0.0f;
    as_gm(dst)[e] = v;
}

// B operand: 4x16 f32 tile of K-major LDS buffer S[row*TPAD + token].
__device__ __forceinline__ v2f loadB(const float* S, int base) {
    v2f b;
    b.x = S[base];
    b.y = S[base + TPAD];
    return b;
}
// A operand: one packed tile, lane-linear 256B block -> single global_load_b64.
__device__ __forceinline__ v2f loadA(gcfloat* base, long tile) {
    return *(gcv2f*)(base + tile * 64);
}

// One layer: h' = (W h + b) * sigmoid(Wg tpc + bg) + Wb tpc ; tangents d' = (W d)*gate,
// then softplus / sigmoid-chain unless LAST.
template<int KTM, int DOUT, bool LAST>
__device__ void run_layer(const float* WPp,  const float* bbp,
                          const float* WgPp, const float* bgp,
                          const float* WbPp,
                          const float* s_tpc, float* s_h, float* s_d0, float* s_d1,
                          int wave, int lane, int lo, int hi,
                          float* outp, long g0)
{
    constexpr int NM = (DOUT >= 32) ? 2 : 1;            // M-tiles per wave
    const int  mt0    = (DOUT >= 32) ? wave * 2 : 0;    // wave's M-tile base
    const int  mw     = mt0 * 16;
    const bool active = (DOUT >= 32) || (wave == 0);

    v8f acc_h[NM][2], acc_d0[NM][2], acc_d1[NM][2];

    if (active) {
        // ---- context matmuls: gate-pre (g) and hyper-bias (hbv) over 33 K-tiles ----
        v8f g[NM][2], hbv[NM][2];
        #pragma unroll
        for (int mt = 0; mt < NM; ++mt)
            #pragma unroll
            for (int t2 = 0; t2 < 2; ++t2) { g[mt][t2] = zero8(); hbv[mt][t2] = zero8(); }

        {
            gcfloat* pg = as_g(WgPp) + (long)mt0 * KT_CTX * 64 + lane * 2;
            gcfloat* pb = as_g(WbPp) + (long)mt0 * KT_CTX * 64 + lane * 2;
            int bidx = 2 * hi * TPAD + lo;
            #pragma clang loop unroll(disable)
            for (int kt = 0; kt < KT_CTX; ++kt) {
                v2f btp0 = loadB(s_tpc, bidx);
                v2f btp1 = loadB(s_tpc, bidx + 16);
                #pragma unroll
                for (int mt = 0; mt < NM; ++mt) {
                    v2f ag = loadA(pg, mt * KT_CTX + kt);
                    v2f ab = loadA(pb, mt * KT_CTX + kt);
                    g[mt][0]   = wmma_f32(ag, btp0, g[mt][0]);
                    g[mt][1]   = wmma_f32(ag, btp1, g[mt][1]);
                    hbv[mt][0] = wmma_f32(ab, btp0, hbv[mt][0]);
                    hbv[mt][1] = wmma_f32(ab, btp1, hbv[mt][1]);
                }
                bidx += 4 * TPAD;
            }
        }
        // gate = sigmoid(g + bg) ; fold main bias: c0 = gate*b + hb  (hnew = gate*a + c0)
        {
            gcfloat* bgg = as_g(bgp);
            gcfloat* bbg = as_g(bbp);
            #pragma unroll
            for (int mt = 0; mt < NM; ++mt)
                #pragma unroll
                for (int r = 0; r < 8; ++r) {
                    int m = mw + mt * 16 + r + 8 * hi;
                    float bgv = ((DOUT % 16 == 0) || m < DOUT) ? bgg[m] : 0.0f;
                    float bv  = ((DOUT % 16 == 0) || m < DOUT) ? bbg[m] : 0.0f;
                    #pragma unroll
                    for (int t2 = 0; t2 < 2; ++t2) {
                        float gv = sigmoidf(g[mt][t2][r] + bgv);
                        g[mt][t2][r]   = gv;
                        hbv[mt][t2][r] = gv * bv + hbv[mt][t2][r];
                    }
                }
        }

        // ---- main matmuls: primal + 2 tangents share the A operand ----
        v8f a[NM][2], d0[NM][2], d1[NM][2];
        #pragma unroll
        for (int mt = 0; mt < NM; ++mt)
            #pragma unroll
            for (int t2 = 0; t2 < 2; ++t2) { a[mt][t2] = zero8(); d0[mt][t2] = zero8(); d1[mt][t2] = zero8(); }

        {
            gcfloat* pw = as_g(WPp) + (long)mt0 * KTM * 64 + lane * 2;
            int bidx = 2 * hi * TPAD + lo;
            #pragma clang loop unroll(disable)
            for (int kt = 0; kt < KTM; ++kt) {
                v2f bh0 = loadB(s_h,  bidx);
                v2f bh1 = loadB(s_h,  bidx + 16);
                v2f b00 = loadB(s_d0, bidx);
                v2f b01 = loadB(s_d0, bidx + 16);
                v2f b10 = loadB(s_d1, bidx);
                v2f b11 = loadB(s_d1, bidx + 16);
                #pragma unroll
                for (int mt = 0; mt < NM; ++mt) {
                    v2f aw = loadA(pw, mt * KTM + kt);
                    a[mt][0]  = wmma_f32(aw, bh0, a[mt][0]);
                    a[mt][1]  = wmma_f32(aw, bh1, a[mt][1]);
                    d0[mt][0] = wmma_f32(aw, b00, d0[mt][0]);
                    d0[mt][1] = wmma_f32(aw, b01, d0[mt][1]);
                    d1[mt][0] = wmma_f32(aw, b10, d1[mt][0]);
                    d1[mt][1] = wmma_f32(aw, b11, d1[mt][1]);
                }
                bidx += 4 * TPAD;
            }
        }

        // ---- combine + activation:  pre = gate*a + c0 ; tangents *= gate ----
        #pragma unroll
        for (int mt = 0; mt < NM; ++mt)
            #pragma unroll
            for (int t2 = 0; t2 < 2; ++t2)
                #pragma unroll
                for (int r = 0; r < 8; ++r) {
                    float gv  = g[mt][t2][r];
                    float pre = a[mt][t2][r] * gv + hbv[mt][t2][r];
                    float t0  = d0[mt][t2][r] * gv;
                    float t1  = d1[mt][t2][r] * gv;
                    if (!LAST) {
                        float sg = sigmoidf(pre);           // softplus'(pre)
                        acc_h [mt][t2][r] = softplusf(pre);
                        acc_d0[mt][t2][r] = sg * t0;
                        acc_d1[mt][t2][r] = sg * t1;
                    } else {
                        acc_h [mt][t2][r] = pre;
                        acc_d0[mt][t2][r] = t0;
                        acc_d1[mt][t2][r] = t1;
                    }
                }
    }

    __syncthreads();   // everyone done reading s_h/s_d0/s_d1

    if (active) {
        if (!LAST) {
            #pragma unroll
            for (int mt = 0; mt < NM; ++mt)
                #pragma unroll
                for (int t2 = 0; t2 < 2; ++t2)
                    #pragma unroll
                    for (int r = 0; r < 8; ++r) {
                        int m   = mw + mt * 16 + r + 8 * hi;
                        int tok = t2 * 16 + lo;
                        s_h [m * TPAD + tok] = acc_h [mt][t2][r];
                        s_d0[m * TPAD + tok] = acc_d0[mt][t2][r];
                        s_d1[m * TPAD + tok] = acc_d1[mt][t2][r];
                    }
        } else if (hi == 0) {
            // wave 0, lanes 0-15: rows M=0,1 of the 16x16 tile are the 2 outputs
            gfloat* out = as_gm(outp);
            #pragma unroll
            for (int t2 = 0; t2 < 2; ++t2) {
                long gt = g0 + t2 * 16 + lo;
                v2f zd;
                zd.x = acc_h[0][t2][0];
                zd.y = acc_h[0][t2][1];
                *(gv2f*)(out + gt * 2) = zd;                                     // z_dot
                out[2L * NTOK + gt] = -(acc_d0[0][t2][0] + acc_d1[0][t2][1]);    // -trace
            }
        }
    }
    __syncthreads();
}

__global__ __launch_bounds__(256, 4)
void ode_jvp_kernel(const float* __restrict__ t, const float* __restrict__ z,
                    const float* __restrict__ cond,
                    const float* WP0, const float* WgP0, const float* WbP0,
                    const float* WP1, const float* WgP1, const float* WbP1,
                    const float* WP2, const float* WgP2, const float* WbP2,
                    const float* WP3, const float* WgP3, const float* WbP3,
                    const float* b0, const float* bg0, const float* b1, const float* bg1,
                    const float* b2, const float* bg2, const float* b3, const float* bg3,
                    float* __restrict__ out)
{
    __shared__ float s_tpc[132 * TPAD];   // context rows: [pos, t, cond(128), 0, 0]
    __shared__ float s_h  [256 * TPAD];   // activations, K-major
    __shared__ float s_d0 [256 * TPAD];   // tangent e0
    __shared__ float s_d1 [256 * TPAD];   // tangent e1

    const int  tid  = threadIdx.x;
    const int  wave = tid >> 5;
    const int  lane = tid & 31;
    const int  lo   = lane & 15;
    const int  hi   = lane >> 4;
    const long g0   = (long)blockIdx.x * TOK;
    const int  bidx = (int)(g0 >> 7);     // batch index (S = 128)
    const int  sb   = (int)(g0 & 127);    // sequence base
    const float tv  = *as_g(t);

    // build tpc = [pos, t, cond, 0, 0] broadcast over 32 tokens
    for (int i = tid; i < 132 * TOK; i += 256) {
        int r = i >> 5, tok = i & 31;
        float v;
        if      (r == 0)  v = (float)(sb + tok + 1) * (1.0f / 128.0f);
        else if (r == 1)  v = tv;
        else if (r < 130) v = as_g(cond)[bidx * 128 + (r - 2)];
        else              v = 0.0f;
        s_tpc[r * TPAD + tok] = v;
    }
    // init h = z (rows 0,1), tangents = e0, e1; rows 2,3 zero-padded for K=4 step
    for (int i = tid; i < 4 * TOK; i += 256) {
        int r = i >> 5, tok = i & 31;
        s_h [r * TPAD + tok] = (r < 2) ? as_g(z)[(g0 + tok) * 2 + r] : 0.0f;
        s_d0[r * TPAD + tok] = (r == 0) ? 1.0f : 0.0f;
        s_d1[r * TPAD + tok] = (r == 1) ? 1.0f : 0.0f;
    }
    // third output: zeros_like(condition), striped across blocks (8192*32 == NTOK)
    if (tid < TOK) as_gm(out)[3L * NTOK + g0 + tid] = 0.0f;
    __syncthreads();

    run_layer<KT_Z, 256, false>(WP0, b0, WgP0, bg0, WbP0, s_tpc, s_h, s_d0, s_d1, wave, lane, lo, hi, out, g0);
    run_layer<KT_H, 256, false>(WP1, b1, WgP1, bg1, WbP1, s_tpc, s_h, s_d0, s_d1, wave, lane, lo, hi, out, g0);
    run_layer<KT_H, 256, false>(WP2, b2, WgP2, bg2, WbP2, s_tpc, s_h, s_d0, s_d1, wave, lane, lo, hi, out, g0);
    run_layer<KT_H,   2, true >(WP3, b3, WgP3, bg3, WbP3, s_tpc, s_h, s_d0, s_d1, wave, lane, lo, hi, out, g0);
}

extern "C" void kernel_launch(void* const* d_in, const int* in_sizes, int n_in,
                              void* d_out, int out_size, void* d_ws, size_t ws_size,
                              hipStream_t stream) {
    (void)in_sizes; (void)n_in; (void)out_size; (void)ws_size;
    const float* t    = (const float*)d_in[0];
    const float* z    = (const float*)d_in[1];
    const float* cond = (const float*)d_in[2];
    // d_in[3 + 5*l + {0:W, 1:b, 2:Wg, 3:bg, 4:Wb}]
    const float* W [4], *bb[4], *Wg[4], *bg[4], *Wb[4];
    for (int l = 0; l < 4; ++l) {
        W [l] = (const float*)d_in[3 + 5 * l + 0];
        bb[l] = (const float*)d_in[3 + 5 * l + 1];
        Wg[l] = (const float*)d_in[3 + 5 * l + 2];
        bg[l] = (const float*)d_in[3 + 5 * l + 3];
        Wb[l] = (const float*)d_in[3 + 5 * l + 4];
    }

    // carve packed-weight buffers out of d_ws
    float* ws = (float*)d_ws;
    size_t off = 0;
    auto carve = [&](int MT, int KT) { float* p = ws + off; off += (size_t)MT * KT * 64; return p; };
    float* WP [4]; float* WgP[4]; float* WbP[4];
    const int dout[4] = {256, 256, 256, 2};
    const int din [4] = {2, 256, 256, 256};
    const int ktm [4] = {KT_Z, KT_H, KT_H, KT_H};
    const int mt  [4] = {MT_H, MT_H, MT_H, MT_O};
    for (int l = 0; l < 4; ++l) {
        WP [l] = carve(mt[l], ktm[l]);
        WgP[l] = carve(mt[l], KT_CTX);
        WbP[l] = carve(mt[l], KT_CTX);
    }
    // repack (zero-padded, WMMA A-operand order)
    auto pack = [&](const float* src, float* dst, int DOUT, int DIN, int MT, int KT) {
        int total = MT * KT * 64;
        pack_kernel<<<(total + 255) / 256, 256, 0, stream>>>(src, dst, DOUT, DIN, KT, total);
    };
    for (int l = 0; l < 4; ++l) {
        pack(W [l], WP [l], dout[l], din[l], mt[l], ktm[l]);
        pack(Wg[l], WgP[l], dout[l], 130,    mt[l], KT_CTX);
        pack(Wb[l], WbP[l], dout[l], 130,    mt[l], KT_CTX);
    }

    dim3 grid((unsigned)(NTOK / TOK));   // 8192 workgroups x 32 tokens
    ode_jvp_kernel<<<grid, 256, 0, stream>>>(
        t, z, cond,
        WP[0], WgP[0], WbP[0],
        WP[1], WgP[1], WbP[1],
        WP[2], WgP[2], WbP[2],
        WP[3], WgP[3], WbP[3],
        bb[0], bg[0], bb[1], bg[1], bb[2], bg[2], bb[3], bg[3],
        (float*)d_out);
}